// NNAttentionHead_72765335929554
// MI455X (gfx1250) — compile-verified
//
#include <hip/hip_runtime.h>
#include <hip/hip_bf16.h>

typedef float v2f __attribute__((ext_vector_type(2)));
typedef float v8f __attribute__((ext_vector_type(8)));

#define B_DIM 2
#define T_DIM 512
#define C_DIM 64
#define H_DIM 64

// ---------------------------------------------------------------------------
// Pass 1: hq[bt,c] = x1 . Wq[c,:], hkb[bt,c] = x1 . Wk[c,:] + b1[c],
//         v[bt,h]  = x[bt,:] . Wv[h,:]     (x1 = [pos[t]; x[bt]], 128 wide)
// ---------------------------------------------------------------------------
__global__ __launch_bounds__(64) void nnah_proj_kernel(
    const float* __restrict__ x, const float* __restrict__ pos,
    const float* __restrict__ W1, const float* __restrict__ b1,
    const float* __restrict__ Wv,
    float* __restrict__ hq, float* __restrict__ hkb, float* __restrict__ v) {
  const int bt = blockIdx.x;          // b*T + t
  const int t  = bt & (T_DIM - 1);
  const int c  = threadIdx.x;         // 0..63

  __shared__ float x1[128];
  x1[c]      = pos[t * C_DIM + c];
  x1[64 + c] = x[bt * C_DIM + c];
  __syncthreads();

  float aq = 0.f, ak = 0.f, av = 0.f;
  const float* wrow = W1 + c * (4 * C_DIM);   // W1 row c, 256 wide
#pragma unroll 8
  for (int d = 0; d < 128; ++d) {
    float xv = x1[d];
    ak = fmaf(xv, wrow[d], ak);                 // Wk = W1[:, :128]
    aq = fmaf(xv, wrow[128 + d], aq);           // Wq = W1[:, 128:]
  }
  const float* vrow = Wv + c * C_DIM;           // Wv row h=c
#pragma unroll 8
  for (int d = 0; d < 64; ++d) av = fmaf(x1[64 + d], vrow[d], av);

  hq[bt * C_DIM + c]  = aq;
  hkb[bt * C_DIM + c] = ak + b1[c];
  v[bt * C_DIM + c]   = av;
}

// ---------------------------------------------------------------------------
// Score tile computation shared by both paths (mapping: ii=tid>>4, jj=jg+16u
// -> conflict-free sK reads, coalesced row-contiguous writes).
// ---------------------------------------------------------------------------
__device__ __forceinline__ float nnah_score_dot(
    const float (*sHq)[68], const float (*sK)[68], const float* sW2,
    int ii, int jj) {
  float acc = 0.f;
#pragma unroll
  for (int cc = 0; cc < 64; cc += 4) {
    float4 q4 = *(const float4*)&sHq[ii][cc];
    float4 k4 = *(const float4*)&sK[jj][cc];
    acc = fmaf(fmaxf(q4.x + k4.x, 0.f), sW2[cc + 0], acc);
    acc = fmaf(fmaxf(q4.y + k4.y, 0.f), sW2[cc + 1], acc);
    acc = fmaf(fmaxf(q4.z + k4.z, 0.f), sW2[cc + 2], acc);
    acc = fmaf(fmaxf(q4.w + k4.w, 0.f), sW2[cc + 3], acc);
  }
  return acc;
}

// ---------------------------------------------------------------------------
// Split pass 2a: one block per (b, 16-row i-tile, 64-col j-tile).
// Uniform work per block -> good load balance across WGPs.
// ---------------------------------------------------------------------------
__global__ __launch_bounds__(256) void nnah_score_kernel(
    const float* __restrict__ hq, const float* __restrict__ hkb,
    const float* __restrict__ W2, const float* __restrict__ b2,
    float* __restrict__ scores) {
  const int blk = blockIdx.x;            // 0..511
  const int b   = blk >> 8;
  const int rem = blk & 255;
  const int it  = rem >> 3;
  const int jt  = rem & 7;
  const int i0  = it * 16;
  const int njt = (i0 + 16 + 63) >> 6;   // causal j-tile count for this i-tile
  if (jt >= njt) return;                 // block-uniform early exit

  __shared__ float sHq[16][68];
  __shared__ float sK[64][68];
  __shared__ float sW2[64];

  const int tid = threadIdx.x;
  {
    int row = tid >> 4;
    int c4  = (tid & 15) << 2;
    *(float4*)&sHq[row][c4] =
        *(const float4*)&hq[((b * T_DIM) + i0 + row) * C_DIM + c4];
  }
#pragma unroll
  for (int q = 0; q < 4; ++q) {
    int f   = tid + q * 256;
    int row = f >> 4;
    int c4  = (f & 15) << 2;
    *(float4*)&sK[row][c4] =
        *(const float4*)&hkb[((b * T_DIM) + jt * 64 + row) * C_DIM + c4];
  }
  if (tid < 64) sW2[tid] = W2[tid];
  const float b2v   = b2[0];
  const float scale = 0.125f;            // C^-0.5, C=64
  __syncthreads();

  const int ii   = tid >> 4;
  const int jg   = tid & 15;
  const int imax = i0 + ii;
  float* orow = scores + (size_t)b * T_DIM * T_DIM + (size_t)(i0 + ii) * T_DIM;
#pragma unroll
  for (int u = 0; u < 4; ++u) {
    int jj = jg + 16 * u;
    int j  = jt * 64 + jj;
    float acc = nnah_score_dot(sHq, sK, sW2, ii, jj);
    orow[j] = (j <= imax) ? (acc + b2v) * scale : -__builtin_inff();
  }
}

// ---------------------------------------------------------------------------
// Split pass 2b: per (b, i-tile): softmax + wei @ v with f32 WMMA.
// ---------------------------------------------------------------------------
__global__ __launch_bounds__(256) void nnah_softmax_av_kernel(
    const float* __restrict__ scores, const float* __restrict__ v,
    float* __restrict__ out) {
  const int blk  = blockIdx.x;           // 0..63
  const int b    = blk >> 5;
  const int it   = blk & 31;
  const int i0   = it * 16;
  const int tid  = threadIdx.x;
  const int lane = tid & 31;
  const int wave = __builtin_amdgcn_readfirstlane(tid >> 5);
  const int njt  = (i0 + 16 + 63) >> 6;
  const int jc   = njt * 64;

  __shared__ float sWei[16][516];        // stride%64==4 -> conflict-free A reads
  __shared__ float sTile[64][68];
  __shared__ float sRed[16][16];
  __shared__ float sRowSum[16];
  __shared__ float sPart[8][256];

  // load score rows from workspace (L2-resident)
  {
    const int nf4row = njt * 16;         // float4 per row
    for (int f = tid; f < 16 * nf4row; f += 256) {
      int row = f / nf4row;
      int c4  = (f - row * nf4row) << 2;
      *(float4*)&sWei[row][c4] = *(const float4*)
          &scores[(size_t)b * T_DIM * T_DIM + (size_t)(i0 + row) * T_DIM + c4];
    }
  }
  __syncthreads();

  // ---- causal softmax per row (unnormalized; defer 1/sum) ----
  {
    const int row = tid >> 4;
    const int lg  = tid & 15;
    float m = -__builtin_inff();
    for (int j = lg; j < jc; j += 16) m = fmaxf(m, sWei[row][j]);
    sRed[row][lg] = m;
    __syncthreads();
    float rmax = sRed[row][0];
#pragma unroll
    for (int k = 1; k < 16; ++k) rmax = fmaxf(rmax, sRed[row][k]);
    __syncthreads();
    float s = 0.f;
    for (int j = lg; j < jc; j += 16) {
      float e = __expf(sWei[row][j] - rmax);   // exp(-inf)=0 handles the mask
      sWei[row][j] = e;
      s += e;
    }
    sRed[row][lg] = s;
    __syncthreads();
    if (lg == 0) {
      float rs = 0.f;
#pragma unroll
      for (int k = 0; k < 16; ++k) rs += sRed[row][k];
      sRowSum[row] = rs;
    }
  }

  // ---- out(16x64) = wei(16xK) @ v(Kx64) via V_WMMA_F32_16X16X4_F32 ----
  const int nt    = wave & 3;
  const int par   = wave >> 2;
  const int h0    = nt * 16;
  const int mrow  = lane & 15;
  const int khalf = lane >> 4;
  v8f acc = {0.f, 0.f, 0.f, 0.f, 0.f, 0.f, 0.f, 0.f};

  for (int kt = 0; kt < njt; ++kt) {
    __syncthreads();
#pragma unroll
    for (int q = 0; q < 4; ++q) {
      int f   = tid + q * 256;
      int row = f >> 4;
      int c4  = (f & 15) << 2;
      *(float4*)&sTile[row][c4] =
          *(const float4*)&v[((b * T_DIM) + kt * 64 + row) * C_DIM + c4];
    }
    __syncthreads();
    if ((kt & 1) == par) {               // scalar -> wave-uniform, EXEC all-1s
#pragma unroll
      for (int k0 = 0; k0 < 64; k0 += 4) {
        int kk = k0 + khalf * 2;
        v2f a, bf;
        a.x  = sWei[mrow][kt * 64 + kk];
        a.y  = sWei[mrow][kt * 64 + kk + 1];
        bf.x = sTile[kk][h0 + mrow];
        bf.y = sTile[kk + 1][h0 + mrow];
        acc = __builtin_amdgcn_wmma_f32_16x16x4_f32(
            false, a, false, bf, (short)0, acc, false, false);
      }
    }
  }

  __syncthreads();
#pragma unroll
  for (int r = 0; r < 8; ++r) {
    int m = khalf * 8 + r;
    sPart[wave][m * 16 + mrow] = acc[r];
  }
  __syncthreads();

  for (int e = tid; e < 16 * H_DIM; e += 256) {
    int m   = e >> 6;
    int h   = e & 63;
    int nt2 = h >> 4;
    int n   = h & 15;
    float val = sPart[nt2][m * 16 + n] + sPart[nt2 + 4][m * 16 + n];
    out[((b * T_DIM) + i0 + m) * H_DIM + h] = val / sRowSum[m];
  }
}

// ---------------------------------------------------------------------------
// Fallback fused pass 2 (used only if workspace too small for score buffer).
// ---------------------------------------------------------------------------
__global__ __launch_bounds__(256) void nnah_attn_fused_kernel(
    const float* __restrict__ hq, const float* __restrict__ hkb,
    const float* __restrict__ v,  const float* __restrict__ W2,
    const float* __restrict__ b2, float* __restrict__ out) {
  const int blk  = blockIdx.x;
  const int b    = blk >> 5;
  const int it   = blk & 31;
  const int i0   = it * 16;
  const int tid  = threadIdx.x;
  const int lane = tid & 31;
  const int wave = __builtin_amdgcn_readfirstlane(tid >> 5);
  const int njt  = (i0 + 16 + 63) >> 6;
  const int jc   = njt * 64;

  __shared__ float sHq[16][68];
  __shared__ float sTile[64][68];
  __shared__ float sWei[16][516];
  __shared__ float sRed[16][16];
  __shared__ float sRowSum[16];
  __shared__ float sW2[64];
  __shared__ float sPart[8][256];

  {
    int row = tid >> 4;
    int c4  = (tid & 15) << 2;
    *(float4*)&sHq[row][c4] =
        *(const float4*)&hq[((b * T_DIM) + i0 + row) * C_DIM + c4];
  }
  if (tid < 64) sW2[tid] = W2[tid];
  const float b2v   = b2[0];
  const float scale = 0.125f;

  const int ii   = tid >> 4;
  const int jg   = tid & 15;
  const int imax = i0 + ii;

  for (int jt = 0; jt < njt; ++jt) {
    __syncthreads();
#pragma unroll
    for (int q = 0; q < 4; ++q) {
      int f   = tid + q * 256;
      int row = f >> 4;
      int c4  = (f & 15) << 2;
      *(float4*)&sTile[row][c4] =
          *(const float4*)&hkb[((b * T_DIM) + jt * 64 + row) * C_DIM + c4];
    }
    __syncthreads();
#pragma unroll
    for (int u = 0; u < 4; ++u) {
      int jj = jg + 16 * u;
      int j  = jt * 64 + jj;
      float acc = nnah_score_dot(sHq, sTile, sW2, ii, jj);
      sWei[ii][j] = (j <= imax) ? (acc + b2v) * scale : -__builtin_inff();
    }
  }
  __syncthreads();

  {
    const int row = tid >> 4;
    const int lg  = tid & 15;
    float m = -__builtin_inff();
    for (int j = lg; j < jc; j += 16) m = fmaxf(m, sWei[row][j]);
    sRed[row][lg] = m;
    __syncthreads();
    float rmax = sRed[row][0];
#pragma unroll
    for (int k = 1; k < 16; ++k) rmax = fmaxf(rmax, sRed[row][k]);
    __syncthreads();
    float s = 0.f;
    for (int j = lg; j < jc; j += 16) {
      float e = __expf(sWei[row][j] - rmax);
      sWei[row][j] = e;
      s += e;
    }
    sRed[row][lg] = s;
    __syncthreads();
    if (lg == 0) {
      float rs = 0.f;
#pragma unroll
      for (int k = 0; k < 16; ++k) rs += sRed[row][k];
      sRowSum[row] = rs;
    }
  }

  const int nt    = wave & 3;
  const int par   = wave >> 2;
  const int h0    = nt * 16;
  const int mrow  = lane & 15;
  const int khalf = lane >> 4;
  v8f acc = {0.f, 0.f, 0.f, 0.f, 0.f, 0.f, 0.f, 0.f};

  for (int kt = 0; kt < njt; ++kt) {
    __syncthreads();
#pragma unroll
    for (int q = 0; q < 4; ++q) {
      int f   = tid + q * 256;
      int row = f >> 4;
      int c4  = (f & 15) << 2;
      *(float4*)&sTile[row][c4] =
          *(const float4*)&v[((b * T_DIM) + kt * 64 + row) * C_DIM + c4];
    }
    __syncthreads();
    if ((kt & 1) == par) {
#pragma unroll
      for (int k0 = 0; k0 < 64; k0 += 4) {
        int kk = k0 + khalf * 2;
        v2f a, bf;
        a.x  = sWei[mrow][kt * 64 + kk];
        a.y  = sWei[mrow][kt * 64 + kk + 1];
        bf.x = sTile[kk][h0 + mrow];
        bf.y = sTile[kk + 1][h0 + mrow];
        acc = __builtin_amdgcn_wmma_f32_16x16x4_f32(
            false, a, false, bf, (short)0, acc, false, false);
      }
    }
  }

  __syncthreads();
#pragma unroll
  for (int r = 0; r < 8; ++r) {
    int m = khalf * 8 + r;
    sPart[wave][m * 16 + mrow] = acc[r];
  }
  __syncthreads();

  for (int e = tid; e < 16 * H_DIM; e += 256) {
    int m   = e >> 6;
    int h   = e & 63;
    int nt2 = h >> 4;
    int n   = h & 15;
    float val = sPart[nt2][m * 16 + n] + sPart[nt2 + 4][m * 16 + n];
    out[((b * T_DIM) + i0 + m) * H_DIM + h] = val / sRowSum[m];
  }
}

// ---------------------------------------------------------------------------
extern "C" void kernel_launch(void* const* d_in, const int* in_sizes, int n_in,
                              void* d_out, int out_size, void* d_ws, size_t ws_size,
                              hipStream_t stream) {
  const float* x   = (const float*)d_in[0];   // (B,T,C)
  const float* pos = (const float*)d_in[1];   // (BS,C)
  const float* W1  = (const float*)d_in[2];   // (C,4C)
  const float* b1  = (const float*)d_in[3];   // (C,)
  const float* W2  = (const float*)d_in[4];   // (1,C)
  const float* b2  = (const float*)d_in[5];   // (1,)
  const float* Wv  = (const float*)d_in[6];   // (H,C)
  float* out = (float*)d_out;                 // (B,T,H)

  const size_t projElems  = (size_t)3 * B_DIM * T_DIM * C_DIM;  // hq,hkb,v
  const size_t scoreElems = (size_t)B_DIM * T_DIM * T_DIM;
  float* hq     = (float*)d_ws;
  float* hkb    = hq + B_DIM * T_DIM * C_DIM;
  float* vws    = hkb + B_DIM * T_DIM * C_DIM;
  float* scores = hq + projElems;

  nnah_proj_kernel<<<B_DIM * T_DIM, 64, 0, stream>>>(x, pos, W1, b1, Wv,
                                                     hq, hkb, vws);

  if (ws_size >= (projElems + scoreElems) * sizeof(float)) {
    // balanced split path: 512 uniform score blocks, then softmax+WMMA
    nnah_score_kernel<<<B_DIM * 256, 256, 0, stream>>>(hq, hkb, W2, b2, scores);
    nnah_softmax_av_kernel<<<B_DIM * (T_DIM / 16), 256, 0, stream>>>(scores, vws,
                                                                     out);
  } else {
    nnah_attn_fused_kernel<<<B_DIM * (T_DIM / 16), 256, 0, stream>>>(
        hq, hkb, vws, W2, b2, out);
  }
}